// DiscreteHMM_14104672600840
// MI455X (gfx1250) — compile-verified
//
#include <hip/hip_runtime.h>

typedef __attribute__((ext_vector_type(2))) float v2f;
typedef __attribute__((ext_vector_type(8))) float v8f;

constexpr int HB = 64;    // batch
constexpr int HT = 4096;  // time
constexpr int HK = 32;    // states
constexpr float LOG2E = 1.44269504088896340736f;
constexpr float LN2   = 0.69314718055994530942f;

// raw hardware transcendentals: v_exp_f32 is exp2, v_log_f32 is log2.
// Inputs here are never denormal (sums of softmax probabilities with the
// max-scaled term == p_max = 1), so the libm denorm fixups are dead weight
// on the serial scan critical path.
#if __has_builtin(__builtin_amdgcn_exp2f)
__device__ __forceinline__ float fexp2(float x){ return __builtin_amdgcn_exp2f(x); }
#else
__device__ __forceinline__ float fexp2(float x){ return exp2f(x); }
#endif
#if __has_builtin(__builtin_amdgcn_logf)
__device__ __forceinline__ float flog2(float x){ return __builtin_amdgcn_logf(x); }
#else
__device__ __forceinline__ float flog2(float x){ return log2f(x); }
#endif

__device__ __forceinline__ float rmax32(float x){
#pragma unroll
  for (int m = 16; m >= 1; m >>= 1) x = fmaxf(x, __shfl_xor(x, m, 32));
  return x;
}
__device__ __forceinline__ float rsum32(float x){
#pragma unroll
  for (int m = 16; m >= 1; m >>= 1) x += __shfl_xor(x, m, 32);
  return x;
}
// reduce across the 16-lane half-groups only (lane bits 0..3)
__device__ __forceinline__ float rmax16(float x){
#pragma unroll
  for (int m = 8; m >= 1; m >>= 1) x = fmaxf(x, __shfl_xor(x, m, 32));
  return x;
}

__device__ __forceinline__ v8f wmma4(v2f a, v2f b, v8f c){
  // D(16x16,f32) = A(16x4,f32) * B(4x16,f32) + C
  return __builtin_amdgcn_wmma_f32_16x16x4_f32(
      /*neg_a=*/false, a, /*neg_b=*/false, b,
      /*c_mod=*/(short)0, c, /*reuse_a=*/false, /*reuse_b=*/false);
}

// emissions for one 16-batch block at time t, in WMMA C/D register layout:
// value (nt, r) on lane = em[b = base + r + 8*h][t][k = nt*16 + ln]
__device__ __forceinline__ void load_em(float (&dst)[2][8], const float* emb,
                                        int t, int h, int ln){
#pragma unroll
  for (int nt = 0; nt < 2; ++nt)
#pragma unroll
    for (int r = 0; r < 8; ++r)
      dst[nt][r] = emb[((size_t)(r + 8*h)*HT + t)*HK + nt*16 + ln];
}

__device__ __forceinline__ void store_tiles(float* dstb, int t, int h, int ln,
                                            const float (&v)[2][8]){
#pragma unroll
  for (int nt = 0; nt < 2; ++nt)
#pragma unroll
    for (int r = 0; r < 8; ++r)
      dstb[((size_t)(r + 8*h)*HT + t)*HK + nt*16 + ln] = v[nt][r];
}

// One recursion step. DIR==0 (forward): p = exp(st - m), st' = log(P) + m + em_t.
// DIR==1 (backward): x = st + em_{t+1}, q = exp(x - m), st' = log(Q) + m.
template<int DIR>
__device__ __forceinline__ void hmm_step(float (&st)[2][8], const float (&emc)[2][8],
                                         const v2f (&breg)[8][2],
                                         float* sPf, const v2f* sP2,
                                         int h, int ln){
  float x[2][8];
#pragma unroll
  for (int nt = 0; nt < 2; ++nt)
#pragma unroll
    for (int r = 0; r < 8; ++r)
      x[nt][r] = (DIR == 0) ? st[nt][r] : st[nt][r] + emc[nt][r];

  float mrow[8];
#pragma unroll
  for (int r = 0; r < 8; ++r) mrow[r] = fmaxf(x[0][r], x[1][r]);
#pragma unroll
  for (int r = 0; r < 8; ++r) mrow[r] = rmax16(mrow[r]);

#pragma unroll
  for (int nt = 0; nt < 2; ++nt)
#pragma unroll
    for (int r = 0; r < 8; ++r)
      sPf[(r + 8*h)*34 + nt*16 + ln] = fexp2((x[nt][r] - mrow[r]) * LOG2E);
  __syncthreads();
  v2f a[8];
#pragma unroll
  for (int kc = 0; kc < 8; ++kc) a[kc] = sP2[ln*17 + 2*kc + h];
  __syncthreads();

  v8f c0 = {0.f,0.f,0.f,0.f,0.f,0.f,0.f,0.f};
  v8f c1 = {0.f,0.f,0.f,0.f,0.f,0.f,0.f,0.f};
#pragma unroll
  for (int kc = 0; kc < 8; ++kc){
    c0 = wmma4(a[kc], breg[kc][0], c0);
    c1 = wmma4(a[kc], breg[kc][1], c1);
  }
#pragma unroll
  for (int r = 0; r < 8; ++r){
    const float b0 = (DIR == 0) ? mrow[r] + emc[0][r] : mrow[r];
    const float b1 = (DIR == 0) ? mrow[r] + emc[1][r] : mrow[r];
    st[0][r] = fmaf(LN2, flog2(c0[r]), b0);
    st[1][r] = fmaf(LN2, flog2(c1[r]), b1);
  }
}

// One wave (32 threads) per block. blockIdx: [0..3] forward over batch block,
// [4..7] backward over batch block. Forward writes alpha into alpha_out
// (== d_out), backward writes beta into beta_out (== d_ws, 32 MB).
__global__ void __launch_bounds__(32)
hmm_scan_kernel(const float* __restrict__ em,
                const float* __restrict__ log_pi,
                const float* __restrict__ log_A,
                float* __restrict__ alpha_out,
                float* __restrict__ beta_out){
  __shared__ float sA[32*33];     // expA row-major, padded
  __shared__ v2f   sP2[16*17];    // p scratch, [M][K] rows of 34 floats (17 float2)
  float* sPf = (float*)sP2;

  const int lane = threadIdx.x;
  const int h    = lane >> 4;     // which 16-lane half
  const int ln   = lane & 15;
  const int dir  = blockIdx.x >> 2;   // 0 = forward, 1 = backward
  const int mb   = blockIdx.x & 3;    // 16-batch block

  // ---- expA = softmax(log_A, axis=1), built once into LDS ----
  for (int i = 0; i < 32; ++i){
    float xv = log_A[i*32 + lane];
    float mx = rmax32(xv);
    float e  = fexp2((xv - mx) * LOG2E);
    float s  = rsum32(e);
    sA[i*33 + lane] = e / s;
  }
  __syncthreads();

  // ---- B operands in WMMA B-layout, resident in VGPRs for the whole scan ----
  // fwd:  B[K=i][N=j] = expA[i][j]      (s[b,j] = sum_i p[b,i] expA[i,j])
  // bwd:  B[K=j][N=i] = expA[i][j]      (s[b,i] = sum_j q[b,j] expA[i,j])
  v2f breg[8][2];
#pragma unroll
  for (int kc = 0; kc < 8; ++kc){
#pragma unroll
    for (int nt = 0; nt < 2; ++nt){
      const int kk = 4*kc + 2*h;
      const int nn = nt*16 + ln;
      if (dir == 0){
        breg[kc][nt].x = sA[kk*33 + nn];
        breg[kc][nt].y = sA[(kk+1)*33 + nn];
      } else {
        breg[kc][nt].x = sA[nn*33 + kk];
        breg[kc][nt].y = sA[nn*33 + kk + 1];
      }
    }
  }

  const float* emb = em + (size_t)(mb*16)*HT*HK;

  if (dir == 0){
    // ---- forward scan ----
    float xp = log_pi[lane];
    float mp = rmax32(xp);
    float sp = rsum32(fexp2((xp - mp) * LOG2E));
    float lp = xp - mp - LN2 * flog2(sp);   // log_softmax(log_pi), per lane=k
    float lpn[2];
#pragma unroll
    for (int nt = 0; nt < 2; ++nt) lpn[nt] = __shfl(lp, nt*16 + ln, 32);

    float* ab = alpha_out + (size_t)(mb*16)*HT*HK;
    float al[2][8];
    float emA[2][8], emB[2][8];
    load_em(emA, emb, 0, h, ln);
#pragma unroll
    for (int nt = 0; nt < 2; ++nt)
#pragma unroll
      for (int r = 0; r < 8; ++r) al[nt][r] = lpn[nt] + emA[nt][r];
    store_tiles(ab, 0, h, ln, al);
    load_em(emA, emb, 1, h, ln);

    for (int t = 1; t + 1 < HT; t += 2){
      load_em(emB, emb, t + 1, h, ln);                    // prefetch
      hmm_step<0>(al, emA, breg, sPf, sP2, h, ln);        // uses em_t
      store_tiles(ab, t, h, ln, al);
      load_em(emA, emb, (t + 2 < HT) ? t + 2 : HT - 1, h, ln);
      hmm_step<0>(al, emB, breg, sPf, sP2, h, ln);        // uses em_{t+1}
      store_tiles(ab, t + 1, h, ln, al);
    }
    // tail: t = HT-1 (HT even -> exactly one left), emA holds em[HT-1]
    hmm_step<0>(al, emA, breg, sPf, sP2, h, ln);
    store_tiles(ab, HT - 1, h, ln, al);
  } else {
    // ---- backward scan ----
    float* bb = beta_out + (size_t)(mb*16)*HT*HK;
    float be[2][8];
#pragma unroll
    for (int nt = 0; nt < 2; ++nt)
#pragma unroll
      for (int r = 0; r < 8; ++r) be[nt][r] = 0.f;
    store_tiles(bb, HT - 1, h, ln, be);
    float emA[2][8], emB[2][8];
    load_em(emA, emb, HT - 1, h, ln);                     // em_{t+1} for t=HT-2

    for (int t = HT - 2; t >= 2; t -= 2){
      load_em(emB, emb, t, h, ln);                        // prefetch em_t
      hmm_step<1>(be, emA, breg, sPf, sP2, h, ln);        // beta_t from em_{t+1}
      store_tiles(bb, t, h, ln, be);
      load_em(emA, emb, t - 1, h, ln);                    // prefetch em_{t-1}
      hmm_step<1>(be, emB, breg, sPf, sP2, h, ln);        // beta_{t-1} from em_t
      store_tiles(bb, t - 1, h, ln, be);
    }
    // tail: t = 0 uses em[1]; emA holds em[1] after the last iteration
    hmm_step<1>(be, emA, breg, sPf, sP2, h, ln);
    store_tiles(bb, 0, h, ln, be);
  }
}

// gamma = (alpha + beta) - logsumexp_k(alpha + beta). One wave per (b,t),
// lane = k. alpha aliases out (in-place): each wave reads then writes only
// its own 32 contiguous elements.
__global__ void __launch_bounds__(256)
hmm_gamma_kernel(const float* alpha, const float* beta, float* out, int npairs){
  const int w    = (blockIdx.x*blockDim.x + threadIdx.x) >> 5;
  const int lane = threadIdx.x & 31;
  if (w >= npairs) return;
  const size_t idx = (size_t)w*32 + lane;
  float g  = alpha[idx] + beta[idx];
  float mx = rmax32(g);
  float s  = rsum32(fexp2((g - mx) * LOG2E));
  out[idx] = g - mx - LN2 * flog2(s);
}

extern "C" void kernel_launch(void* const* d_in, const int* in_sizes, int n_in,
                              void* d_out, int out_size, void* d_ws, size_t ws_size,
                              hipStream_t stream) {
  const float* em     = (const float*)d_in[0];  // [B,T,K]
  const float* log_pi = (const float*)d_in[1];  // [K]
  const float* log_A  = (const float*)d_in[2];  // [K,K]
  float* out  = (float*)d_out;                  // alpha then gamma, [B,T,K]
  float* beta = (float*)d_ws;                   // 32 MB scratch

  // 8 single-wave blocks: 4 batch-blocks x {forward, backward}
  hmm_scan_kernel<<<dim3(8), dim3(32), 0, stream>>>(em, log_pi, log_A, out, beta);

  const int npairs = HB * HT;                   // 262144 (b,t) waves
  const int threads = 256;
  const int blocks = (npairs*32 + threads - 1) / threads;
  hmm_gamma_kernel<<<dim3(blocks), dim3(threads), 0, stream>>>(out, beta, out, npairs);
}